// MambaBlock_52948356825272
// MI455X (gfx1250) — compile-verified
//
#include <hip/hip_runtime.h>
#include <math.h>

// Problem constants (from reference): B=2, L=1024, IN=12, E=24, N=24, dt_rank=4, K=4
#define NB     2
#define LSEQ   1024
#define IN_DIM 12
#define EDIM   24
#define NS     24
#define DTR    4
#define KW     4
#define SELW   (2*NS + DTR)   // 52
#define LS     33             // LDS row stride (odd -> conflict free)

typedef __attribute__((ext_vector_type(2))) float v2f;
typedef __attribute__((ext_vector_type(8))) float v8f;

__device__ __forceinline__ float silu_f(float v)     { return v / (1.f + __expf(-v)); }
__device__ __forceinline__ float softplus_f(float v) { return (v > 20.f) ? v : log1pf(__expf(v)); }

// ---------------------------------------------------------------------------
// 32x32 f32 product core (no barriers):  Dm = Am*Bm + C
// where C = Cm (read in C-layout from LDS) if USE_C, else diagc*I.
// V_WMMA_F32_16X16X4_F32 as 2x2 tiles of 16x16, K-loop of 8 (K=4).
// ISA layouts (CDNA5 ISA):
//   A 16x4 : lane m=lane&15, VGPR0 = K = (lane<16 ? 0 : 2), VGPR1 = K+1
//   B 4x16 : lane n=lane&15, VGPR0 = row K = (lane<16 ? 0 : 2), VGPR1 = K+1
//   C 16x16: VGPR v -> M = v + (lane>=16 ? 8 : 0), N = lane&15
// Caller is responsible for __syncthreads() fencing around regions.
// ---------------------------------------------------------------------------
template<bool USE_C>
__device__ __forceinline__ void mm32_core(const float* Am, const float* Bm,
                                          const float* Cm, float diagc,
                                          float* Dm, int lane)
{
    const int mn    = lane & 15;
    const int khalf = (lane >> 4) << 1;   // 0 or 2
    const int mofs  = (lane >> 4) << 3;   // 0 or 8
#pragma unroll
    for (int ti = 0; ti < 2; ++ti) {
#pragma unroll
        for (int tj = 0; tj < 2; ++tj) {
            v8f c;
#pragma unroll
            for (int v = 0; v < 8; ++v) {
                float cv;
                if (USE_C) cv = Cm[(ti*16 + mofs + v)*LS + tj*16 + mn];
                else       cv = (ti == tj && (mofs + v) == mn) ? diagc : 0.f;
                c[v] = cv;
            }
#pragma unroll
            for (int kk = 0; kk < 8; ++kk) {
                const int k0 = kk*4 + khalf;
                v2f a, b;
                a[0] = Am[(ti*16 + mn)*LS + k0];
                a[1] = Am[(ti*16 + mn)*LS + k0 + 1];
                b[0] = Bm[(k0    )*LS + tj*16 + mn];
                b[1] = Bm[(k0 + 1)*LS + tj*16 + mn];
                c = __builtin_amdgcn_wmma_f32_16x16x4_f32(
                        /*neg_a=*/false, a, /*neg_b=*/false, b,
                        /*c_mod=*/(short)0, c, /*reuse_a=*/false, /*reuse_b=*/false);
            }
#pragma unroll
            for (int v = 0; v < 8; ++v)
                Dm[(ti*16 + mofs + v)*LS + tj*16 + mn] = c[v];
        }
    }
}

// ---------------------------------------------------------------------------
// K1: xz = x @ W_in  -> xs_pre (first E cols), z (last E cols)
// ---------------------------------------------------------------------------
__global__ void k_inproj(const float* __restrict__ x, const float* __restrict__ Win,
                         float* __restrict__ xs_pre, float* __restrict__ z)
{
    int t = blockIdx.x*blockDim.x + threadIdx.x;
    const int total = NB*LSEQ*(2*EDIM);
    if (t >= total) return;
    int c  = t % (2*EDIM);
    int bl = t / (2*EDIM);
    float acc = 0.f;
#pragma unroll
    for (int i = 0; i < IN_DIM; ++i)
        acc += x[bl*IN_DIM + i] * Win[i*(2*EDIM) + c];
    if (c < EDIM) xs_pre[bl*EDIM + c] = acc;
    else          z[bl*EDIM + (c - EDIM)] = acc;
}

// ---------------------------------------------------------------------------
// K2: depthwise causal conv (K=4) + bias + SiLU
// ---------------------------------------------------------------------------
__global__ void k_conv(const float* __restrict__ xs_pre, const float* __restrict__ cw,
                       const float* __restrict__ cb, float* __restrict__ xs)
{
    int t = blockIdx.x*blockDim.x + threadIdx.x;
    const int total = NB*LSEQ*EDIM;
    if (t >= total) return;
    int e = t % EDIM;
    int l = (t / EDIM) % LSEQ;
    int b = t / (EDIM*LSEQ);
    float acc = cb[e];
#pragma unroll
    for (int k = 0; k < KW; ++k) {
        int ls = l - (KW-1) + k;
        if (ls >= 0)
            acc += xs_pre[((b*LSEQ + ls)*EDIM) + e] * cw[e*KW + k];
    }
    xs[t] = silu_f(acc);
}

// ---------------------------------------------------------------------------
// K3: sel = xs @ W_sel -> B_t, C_t, dt_pre
// ---------------------------------------------------------------------------
__global__ void k_sel(const float* __restrict__ xs, const float* __restrict__ Wsel,
                      float* __restrict__ Bt, float* __restrict__ Ct,
                      float* __restrict__ dtp)
{
    int t = blockIdx.x*blockDim.x + threadIdx.x;
    const int total = NB*LSEQ*SELW;
    if (t >= total) return;
    int c  = t % SELW;
    int bl = t / SELW;
    float acc = 0.f;
#pragma unroll
    for (int e = 0; e < EDIM; ++e)
        acc += xs[bl*EDIM + e] * Wsel[e*SELW + c];
    if (c < NS)          Bt[bl*NS + c] = acc;
    else if (c < 2*NS)   Ct[bl*NS + (c - NS)] = acc;
    else                 dtp[bl*DTR + (c - 2*NS)] = acc;
}

// ---------------------------------------------------------------------------
// K4: dt = softplus(dt_pre @ W_dt + b_dt) ; u = dt * B_t (elementwise, E==N)
// ---------------------------------------------------------------------------
__global__ void k_dt(const float* __restrict__ dtp, const float* __restrict__ Wdt,
                     const float* __restrict__ bdt, const float* __restrict__ Bt,
                     float* __restrict__ dt, float* __restrict__ u)
{
    int t = blockIdx.x*blockDim.x + threadIdx.x;
    const int total = NB*LSEQ*EDIM;
    if (t >= total) return;
    int e  = t % EDIM;
    int bl = t / EDIM;
    float acc = bdt[e];
#pragma unroll
    for (int r = 0; r < DTR; ++r)
        acc += dtp[bl*DTR + r] * Wdt[r*EDIM + e];
    float d = softplus_f(acc);
    dt[t] = d;
    u[t]  = d * Bt[bl*NS + e];
}

// ---------------------------------------------------------------------------
// K5: per (b,l,e): E = exp(dt*A), phi1 = (dt*A)^-1 (E - I), via joint
// scaling & squaring with the diagonal/accumulator folded into WMMA C:
//   Taylor deg 8 Horner on X/2^j :   T <- X*T + c_k*I        (7 matmuls each)
//   squaring (un-halved):            Q <- Q*E + Q ; E <- E*E (2 matmuls/step,
//                                    one barrier region, independent products)
//   phi1 = Q * 2^-j  (applied in the final matvec)
//   Bx = (phi1 @ (dt*B_t)) * xs.   One wave per instance.
// ---------------------------------------------------------------------------
__global__ void __launch_bounds__(32)
k_expm(const float* __restrict__ Aglob, const float* __restrict__ dt,
       const float* __restrict__ u, const float* __restrict__ xs,
       float* __restrict__ Ad, float* __restrict__ Bx)
{
    __shared__ float buf0[32*LS], buf1[32*LS], buf2[32*LS], buf3[32*LS];
    const int lane = threadIdx.x;
    const int idx  = blockIdx.x;          // (b*L + l)*E + e
    const int e    = idx % EDIM;
    const int bl   = idx / EDIM;
    const float s  = dt[bl*EDIM + e];

    float* sX = buf0;

    // X = s*A, zero-padded to 32x32 (exact: block-diagonal exponentiates blockwise)
    for (int r = 0; r < 32; ++r) {
        float v = 0.f;
        if (r < EDIM && lane < EDIM) v = s * Aglob[r*EDIM + lane];
        sX[r*LS + lane] = v;
    }
    __syncthreads();

    // inf-norm (max abs row sum), wave-reduced -> uniform j
    float rs = 0.f;
    for (int c = 0; c < 32; ++c) rs += fabsf(sX[lane*LS + c]);
#pragma unroll
    for (int off = 16; off >= 1; off >>= 1)
        rs = fmaxf(rs, __shfl_xor(rs, off, 32));
    int j = 0;
    if (rs > 0.25f) j = (int)ceilf(log2f(rs * 4.0f));
    if (j < 0)  j = 0;
    if (j > 15) j = 15;
    const float sc = ldexpf(1.0f, -j);
    for (int r = 0; r < 32; ++r) sX[r*LS + lane] *= sc;
    __syncthreads();

    const float cE[9] = {1.f, 1.f, 0.5f, 1.f/6.f, 1.f/24.f, 1.f/120.f,
                         1.f/720.f, 1.f/5040.f, 1.f/40320.f};
    const float cP[9] = {1.f, 0.5f, 1.f/6.f, 1.f/24.f, 1.f/120.f, 1.f/720.f,
                         1.f/5040.f, 1.f/40320.f, 1.f/362880.f};

    // --- Horner for E = exp(X): first step is elementwise (c8*X + c7*I) ---
    for (int r = 0; r < 32; ++r)
        buf1[r*LS + lane] = cE[8]*sX[r*LS + lane] + ((r == lane) ? cE[7] : 0.f);
    __syncthreads();
    float* pe = buf1;  float* pt = buf2;
#pragma unroll
    for (int k = 6; k >= 0; --k) {
        mm32_core<false>(sX, pe, nullptr, cE[k], pt, lane);
        __syncthreads();
        float* tmp = pe; pe = pt; pt = tmp;     // 7 iters: pe ends in buf2
    }

    // --- Horner for Q = phi1(X): pair (buf3, buf1-free) ---
    for (int r = 0; r < 32; ++r)
        buf3[r*LS + lane] = cP[8]*sX[r*LS + lane] + ((r == lane) ? cP[7] : 0.f);
    __syncthreads();
    float* pp = buf3;  float* qt = pt;          // pt is the free one of {buf1,buf2}
#pragma unroll
    for (int k = 6; k >= 0; --k) {
        mm32_core<false>(sX, pp, nullptr, cP[k], qt, lane);
        __syncthreads();
        float* tmp = pp; pp = qt; qt = tmp;
    }

    // --- Joint squaring: Q <- Q*E + Q ; E <- E*E  (independent, one region) ---
    float* fa = sX;   // X is dead now
    float* fb = qt;
    for (int t = 0; t < j; ++t) {
        mm32_core<true >(pp, pe, pp, 0.f, fa, lane);   // Q_new = Q*E + Q
        mm32_core<false>(pe, pe, nullptr, 0.f, fb, lane); // E_new = E*E
        __syncthreads();
        float* op = pp; float* oe = pe;
        pp = fa; pe = fb; fa = op; fb = oe;
    }

    // --- Store Ad (24x24 corner of E) ---
    float* AdOut = Ad + (size_t)idx * (NS*NS);
    if (lane < NS)
        for (int r = 0; r < NS; ++r)
            AdOut[r*NS + lane] = pe[r*LS + lane];

    // --- Bx[n] = xs * 2^-j * sum_m Q[n][m] * u[m] ---
    if (lane < NS) fa[lane] = u[bl*NS + lane];
    __syncthreads();
    if (lane < NS) {
        float acc = 0.f;
#pragma unroll
        for (int m = 0; m < NS; ++m) acc += pp[lane*LS + m] * fa[m];
        Bx[(size_t)idx*NS + lane] = acc * sc * xs[bl*EDIM + e];
    }
}

// ---------------------------------------------------------------------------
// K6: sequential scan over L, one wave per (b,e) chain; h staged via LDS,
//     next-step Ad prefetched (global_prefetch_b8). Fuses y = h.C_t + D*xs.
// ---------------------------------------------------------------------------
__global__ void __launch_bounds__(32)
k_scan(const float* __restrict__ Ad, const float* __restrict__ Bx,
       const float* __restrict__ Ct, const float* __restrict__ Dvec,
       const float* __restrict__ xs, float* __restrict__ y)
{
    __shared__ float hsh[32];
    const int lane = threadIdx.x;
    const int b = blockIdx.x / EDIM;
    const int e = blockIdx.x % EDIM;
    const float De = Dvec[e];
    float h = 0.f;
    for (int l = 0; l < LSEQ; ++l) {
        const size_t idx = ((size_t)(b*LSEQ + l)*EDIM + e);
        const float* Arow = Ad + idx*(NS*NS);
        if (l + 1 < LSEQ)
            __builtin_prefetch(Ad + (idx + EDIM)*(NS*NS) + lane*18, 0, 1);
        hsh[lane] = h;
        __syncthreads();
        float hn = 0.f;
        if (lane < NS) {
            const float* row = Arow + lane*NS;
            float acc = Bx[idx*NS + lane];
#pragma unroll
            for (int m = 0; m < NS; ++m) acc += row[m] * hsh[m];
            hn = acc;
        }
        __syncthreads();
        h = hn;
        const int bl = b*LSEQ + l;
        float py = (lane < NS) ? h * Ct[bl*EDIM + lane] : 0.f;
#pragma unroll
        for (int off = 16; off >= 1; off >>= 1) py += __shfl_xor(py, off, 32);
        if (lane == 0) y[bl*EDIM + e] = py + De * xs[bl*EDIM + e];
    }
}

// ---------------------------------------------------------------------------
// K7: out = (y * silu(z)) @ W_out
// ---------------------------------------------------------------------------
__global__ void k_out(const float* __restrict__ y, const float* __restrict__ z,
                      const float* __restrict__ Wout, float* __restrict__ out)
{
    int t = blockIdx.x*blockDim.x + threadIdx.x;
    const int total = NB*LSEQ*IN_DIM;
    if (t >= total) return;
    int i  = t % IN_DIM;
    int bl = t / IN_DIM;
    float acc = 0.f;
#pragma unroll
    for (int e = 0; e < EDIM; ++e)
        acc += y[bl*EDIM + e] * silu_f(z[bl*EDIM + e]) * Wout[e*IN_DIM + i];
    out[t] = acc;
}

// ---------------------------------------------------------------------------
extern "C" void kernel_launch(void* const* d_in, const int* in_sizes, int n_in,
                              void* d_out, int out_size, void* d_ws, size_t ws_size,
                              hipStream_t stream)
{
    const float* x     = (const float*)d_in[0];
    const float* Win   = (const float*)d_in[1];
    const float* convw = (const float*)d_in[2];
    const float* convb = (const float*)d_in[3];
    const float* Wsel  = (const float*)d_in[4];
    const float* Wdt   = (const float*)d_in[5];
    const float* bdt   = (const float*)d_in[6];
    const float* Ag    = (const float*)d_in[7];
    const float* Dg    = (const float*)d_in[8];
    const float* Wout  = (const float*)d_in[9];
    float* out = (float*)d_out;

    const size_t BLE = (size_t)NB*LSEQ*EDIM;     // 49152
    float* ws = (float*)d_ws;
    size_t o = 0;
    float* xs_pre = ws + o; o += BLE;
    float* zbuf   = ws + o; o += BLE;
    float* xsb    = ws + o; o += BLE;
    float* Btb    = ws + o; o += BLE;
    float* Ctb    = ws + o; o += BLE;
    float* dtpb   = ws + o; o += (size_t)NB*LSEQ*DTR;
    float* dtb    = ws + o; o += BLE;
    float* ub     = ws + o; o += BLE;
    float* yb     = ws + o; o += BLE;
    float* Bxb    = ws + o; o += BLE*NS;         // 4.7 MB
    float* Adb    = ws + o; o += BLE*NS*NS;      // 113 MB

    const int th = 256;
    {
        int n = NB*LSEQ*(2*EDIM);
        k_inproj<<<(n + th - 1)/th, th, 0, stream>>>(x, Win, xs_pre, zbuf);
    }
    {
        int n = NB*LSEQ*EDIM;
        k_conv<<<(n + th - 1)/th, th, 0, stream>>>(xs_pre, convw, convb, xsb);
    }
    {
        int n = NB*LSEQ*SELW;
        k_sel<<<(n + th - 1)/th, th, 0, stream>>>(xsb, Wsel, Btb, Ctb, dtpb);
    }
    {
        int n = NB*LSEQ*EDIM;
        k_dt<<<(n + th - 1)/th, th, 0, stream>>>(dtpb, Wdt, bdt, Btb, dtb, ub);
    }
    // dominant kernel: 49152 waves, each (14 + 2j) WMMA 32x32 matmuls
    k_expm<<<NB*LSEQ*EDIM, 32, 0, stream>>>(Ag, dtb, ub, xsb, Adb, Bxb);
    // sequential scan: 48 chains
    k_scan<<<NB*EDIM, 32, 0, stream>>>(Adb, Bxb, Ctb, Dg, xsb, yb);
    {
        int n = NB*LSEQ*IN_DIM;
        k_out<<<(n + th - 1)/th, th, 0, stream>>>(yb, zbuf, Wout, out);
    }
}